// LF_21345987461567
// MI455X (gfx1250) — compile-verified
//
#include <hip/hip_runtime.h>
#include <math.h>
#include <stdint.h>

// ---------------------------------------------------------------------------
// Biquad lowpass over 640 independent rows of 96000 fp32 samples.
// Memory-bound: 0.49 GB round trip -> ~21us floor at 23.3 TB/s.
// Strategy: overlap-discard chunking (pole radius 0.414 -> 32-sample warmup
// reduces state error to ~5.6e-13, far below fp32 ulp), staged through LDS
// with CDNA5 async global<->LDS copies for full coalescing, padded LDS
// addressing for conflict-free stride-64 per-thread scans. Async transfers
// use non-temporal cache hints: the data is streamed once (input reuse is
// only the 32-sample warmup overlap, output reuse is zero), so polluting
// the 192MB L2 with dead lines is avoided.
// ---------------------------------------------------------------------------

#define TPB   128                 // threads per workgroup (4 waves of 32)
#define CH    64                  // output samples per thread-chunk
#define WARM  32                  // warm-up samples (0.414^32 ~ 5.6e-13)
#define SPAN  (TPB * CH)          // 8192 samples of output per workgroup
#define MAXSTREAM (SPAN + WARM)   // 8224 samples staged per workgroup

// LDS padding: +2 floats per 64 => per-chunk stride 66 => bank stride 2
// => conflict-free across 32 lanes; keeps b64 (8B) alignment for async ops.
__device__ __forceinline__ int pad_idx(int p) { return p + 2 * (p >> 6); }

#define IN_LDS  (MAXSTREAM + 2 * ((MAXSTREAM + 63) / 64))   // 8482 floats
#define OUT_LDS (SPAN      + 2 * ((SPAN      + 63) / 64))   // 8448 floats

__global__ __launch_bounds__(TPB)
void LF_biquad_kernel(const float* __restrict__ x, float* __restrict__ y,
                      int seg_len, int chunks_per_row,
                      float b0, float b1, float b2, float a1, float a2)
{
    __shared__ __align__(16) float s_in[IN_LDS];
    __shared__ __align__(16) float s_out[OUT_LDS];

    const int row = blockIdx.y;          // 0..639 (B*SEG independent rows)
    const int blk = blockIdx.x;          // block of 128 chunks within the row
    const int tid = threadIdx.x;

    const long long row_off   = (long long)row * (long long)seg_len;
    const int block_start     = blk * SPAN;                        // sample offset in row
    const int active          = min(TPB, chunks_per_row - blk * TPB);
    const int span_act        = active * CH;                       // outputs this WG
    const int w_eff           = min(WARM, block_start);            // warmup available before block
    const int nstream         = w_eff + span_act;                  // floats staged

    // ---- async copy: global -> LDS, coalesced b64 per lane, non-temporal ---
    {
        const float* gsrc = x + row_off + (long long)(block_start - w_eff);
        for (int p = tid * 2; p < nstream; p += TPB * 2) {
            uint32_t la = (uint32_t)(uintptr_t)(&s_in[pad_idx(p)]);
            uint64_t ga = (uint64_t)(uintptr_t)(gsrc + p);
            asm volatile("global_load_async_to_lds_b64 %0, %1, off th:TH_LOAD_NT"
                         :: "v"(la), "v"(ga) : "memory");
        }
        asm volatile("s_wait_asynccnt 0" ::: "memory");
    }
    __syncthreads();

    // ---- per-thread exact IIR scan over warmup + chunk ---------------------
    if (tid < active) {
        const int w      = min(WARM, block_start + tid * CH); // 0 only at row start
        const int base_p = w_eff + tid * CH - w;              // stream pos of first read
        float x1 = 0.f, x2 = 0.f, y1 = 0.f, y2 = 0.f;

        // warm-up: run recurrence, discard outputs
        #pragma unroll 4
        for (int t = 0; t < w; ++t) {
            float x0 = s_in[pad_idx(base_p + t)];
            float f  = fmaf(b0, x0, fmaf(b1, x1, b2 * x2));
            float yv = fmaf(-a1, y1, fmaf(-a2, y2, f));
            x2 = x1; x1 = x0; y2 = y1; y1 = yv;
        }
        // live region: recurrence uses unclamped state, stored value clamped
        int out_q = tid * CH;
        #pragma unroll 4
        for (int t = 0; t < CH; ++t) {
            float x0 = s_in[pad_idx(base_p + w + t)];
            float f  = fmaf(b0, x0, fmaf(b1, x1, b2 * x2));
            float yv = fmaf(-a1, y1, fmaf(-a2, y2, f));
            s_out[pad_idx(out_q + t)] = fminf(fmaxf(yv, -1.f), 1.f);
            x2 = x1; x1 = x0; y2 = y1; y1 = yv;
        }
    }
    __syncthreads();

    // ---- async copy: LDS -> global, coalesced b64, non-temporal store ------
    {
        float* gdst = y + row_off + (long long)block_start;
        for (int q = tid * 2; q < span_act; q += TPB * 2) {
            uint32_t la = (uint32_t)(uintptr_t)(&s_out[pad_idx(q)]);
            uint64_t ga = (uint64_t)(uintptr_t)(gdst + q);
            asm volatile("global_store_async_from_lds_b64 %0, %1, off th:TH_STORE_NT"
                         :: "v"(ga), "v"(la) : "memory");
        }
        asm volatile("s_wait_asynccnt 0" ::: "memory");
    }
}

extern "C" void kernel_launch(void* const* d_in, const int* in_sizes, int n_in,
                              void* d_out, int out_size, void* d_ws, size_t ws_size,
                              hipStream_t stream) {
    (void)n_in; (void)d_ws; (void)ws_size; (void)out_size;

    const float* x = (const float*)d_in[0];
    float* out     = (float*)d_out;

    // Shapes per reference: [B, 960000] fp32, SEG=10 contiguous segments.
    const int T       = 960000;
    const int SEGN    = 10;
    const int Bn      = in_sizes[0] / T;          // 64
    const int seg_len = T / SEGN;                 // 96000
    const int rows    = Bn * SEGN;                // 640 independent rows

    const int chunks_per_row = (seg_len + CH - 1) / CH;          // 1500
    const int blocks_per_row = (chunks_per_row + TPB - 1) / TPB; // 12

    // torchaudio lowpass_biquad coefficients, computed in double as reference does.
    const double PI = 3.14159265358979323846;
    const double w0    = 2.0 * PI * 4000.0 / 16000.0;
    const double alpha = sin(w0) / (2.0 * 0.7071067811865476);
    const double cw    = cos(w0);
    const double b0d = (1.0 - cw) / 2.0;
    const double b1d = (1.0 - cw);
    const double b2d = b0d;
    const double a0d = 1.0 + alpha;
    const double a1d = -2.0 * cw;
    const double a2d = 1.0 - alpha;

    const float b0 = (float)(b0d / a0d);
    const float b1 = (float)(b1d / a0d);
    const float b2 = (float)(b2d / a0d);
    const float a1 = (float)(a1d / a0d);
    const float a2 = (float)(a2d / a0d);

    dim3 grid(blocks_per_row, rows);
    LF_biquad_kernel<<<grid, TPB, 0, stream>>>(x, out, seg_len, chunks_per_row,
                                               b0, b1, b2, a1, a2);
}